// BOW_model_5815385719098
// MI455X (gfx1250) — compile-verified
//
#include <hip/hip_runtime.h>
#include <math.h>

#define B_BAGS 4096
#define D_DIM  512
#define BN_EPS 1e-5f

typedef __attribute__((ext_vector_type(2))) float v2f;
typedef __attribute__((ext_vector_type(8))) float v8f;

// ---------------------------------------------------------------------------
// 1) EmbeddingBag(mean): one block per bag. segment_ids is sorted, so the
//    bag's tokens are a contiguous range found by binary search (uniform
//    across the block -> scalarized by the compiler). 128 threads stream each
//    2KB embedding row as float4 (fully coalesced), accumulate in registers.
// ---------------------------------------------------------------------------
__global__ __launch_bounds__(128) void bow_embed_bag_kernel(
    const int* __restrict__ token_ids, const int* __restrict__ seg,
    const float* __restrict__ emb, float* __restrict__ bow, int T)
{
    const int b   = blockIdx.x;
    const int tid = threadIdx.x;

    // first index with seg[idx] >= b
    int lo = 0, hi = T;
    while (lo < hi) { int mid = (lo + hi) >> 1; if (seg[mid] < b) lo = mid + 1; else hi = mid; }
    const int start = lo;
    // first index with seg[idx] >= b+1
    hi = T;
    while (lo < hi) { int mid = (lo + hi) >> 1; if (seg[mid] < b + 1) lo = mid + 1; else hi = mid; }
    const int end = lo;

    const int   cnt = end - start;
    const float inv = 1.0f / (float)(cnt > 0 ? cnt : 1);

    float4 s = make_float4(0.f, 0.f, 0.f, 0.f);
    for (int t = start; t < end; ++t) {
        const int tok = token_ids[t];
        if (t + 1 < end) {
            const int nt = token_ids[t + 1];
            __builtin_prefetch(emb + (size_t)nt * D_DIM + tid * 4, 0, 0);  // global_prefetch_b8
        }
        const float4 v = ((const float4*)(emb + (size_t)tok * D_DIM))[tid];
        s.x += v.x; s.y += v.y; s.z += v.z; s.w += v.w;
    }
    float4 r; r.x = s.x * inv; r.y = s.y * inv; r.z = s.z * inv; r.w = s.w * inv;
    ((float4*)(bow + (size_t)b * D_DIM))[tid] = r;
}

// ---------------------------------------------------------------------------
// 2) h = bow @ W1^T + b1 via V_WMMA_F32_16X16X4_F32 (fp32 matrix pipe, exact
//    reference precision). Each wave computes a 16x64 tile (4 accumulators,
//    A fragment reused 4x). K=512 in steps of 4.
//    A layout (16x4 f32):  lane (l=lane&15, hf=lane>>4) holds
//        {A[m0+l][k+2hf], A[m0+l][k+2hf+1]}
//    B layout mirrored with N on lanes: {B[k+2hf][n0+l], B[k+2hf+1][n0+l]}
//    where B[k][n] = W1[n][k]  ->  contiguous v2f from W1 row n.
//    C/D layout: acc[r] of lane -> h[m0 + r + 8*hf][n0 + l].
// ---------------------------------------------------------------------------
__global__ __launch_bounds__(256) void bow_gemm_wmma_kernel(
    const float* __restrict__ bow, const float* __restrict__ W1,
    const float* __restrict__ b1, float* __restrict__ h)
{
    const int wid  = blockIdx.x * 8 + (threadIdx.x >> 5);   // 2048 waves total
    const int lane = threadIdx.x & 31;
    const int mt   = wid >> 3;          // 0..255 : M tile (16 rows)
    const int ns   = wid & 7;           // 0..7   : N super-tile (64 cols)
    const int m0   = mt * 16;
    const int nb   = ns * 64;
    const int hf   = lane >> 4;
    const int l    = lane & 15;

    const float* arow = bow + (size_t)(m0 + l) * D_DIM + 2 * hf;
    const float* w0   = W1  + (size_t)(nb +  0 + l) * D_DIM + 2 * hf;
    const float* w1   = W1  + (size_t)(nb + 16 + l) * D_DIM + 2 * hf;
    const float* w2   = W1  + (size_t)(nb + 32 + l) * D_DIM + 2 * hf;
    const float* w3   = W1  + (size_t)(nb + 48 + l) * D_DIM + 2 * hf;

    v8f acc0 = {}, acc1 = {}, acc2 = {}, acc3 = {};

    for (int k = 0; k < D_DIM; k += 4) {
        const v2f a  = *(const v2f*)(arow + k);
        const v2f b0 = *(const v2f*)(w0 + k);
        const v2f b1v = *(const v2f*)(w1 + k);
        const v2f b2v = *(const v2f*)(w2 + k);
        const v2f b3v = *(const v2f*)(w3 + k);
        acc0 = __builtin_amdgcn_wmma_f32_16x16x4_f32(false, a, false, b0,  (short)0, acc0, false, false);
        acc1 = __builtin_amdgcn_wmma_f32_16x16x4_f32(false, a, false, b1v, (short)0, acc1, false, false);
        acc2 = __builtin_amdgcn_wmma_f32_16x16x4_f32(false, a, false, b2v, (short)0, acc2, false, false);
        acc3 = __builtin_amdgcn_wmma_f32_16x16x4_f32(false, a, false, b3v, (short)0, acc3, false, false);
    }

    #pragma unroll
    for (int r = 0; r < 8; ++r) {
        const int row = m0 + r + 8 * hf;
        float* hr = h + (size_t)row * D_DIM;
        hr[nb +  0 + l] = acc0[r] + b1[nb +  0 + l];
        hr[nb + 16 + l] = acc1[r] + b1[nb + 16 + l];
        hr[nb + 32 + l] = acc2[r] + b1[nb + 32 + l];
        hr[nb + 48 + l] = acc3[r] + b1[nb + 48 + l];
    }
}

// ---------------------------------------------------------------------------
// 3) BatchNorm training stats (biased var over B rows), folded to scale/shift.
//    One block per feature column.
// ---------------------------------------------------------------------------
__global__ __launch_bounds__(256) void bow_bn_stats_kernel(
    const float* __restrict__ h, const float* __restrict__ gamma,
    const float* __restrict__ beta, float* __restrict__ scale,
    float* __restrict__ shift)
{
    __shared__ float ssum[256];
    __shared__ float ssq[256];
    const int d = blockIdx.x, tid = threadIdx.x;
    float s = 0.f, q = 0.f;
    for (int r = tid; r < B_BAGS; r += 256) {
        const float x = h[(size_t)r * D_DIM + d];
        s += x; q += x * x;
    }
    ssum[tid] = s; ssq[tid] = q; __syncthreads();
    for (int off = 128; off; off >>= 1) {
        if (tid < off) { ssum[tid] += ssum[tid + off]; ssq[tid] += ssq[tid + off]; }
        __syncthreads();
    }
    if (tid == 0) {
        const float mean = ssum[0] * (1.0f / B_BAGS);
        const float var  = ssq[0] * (1.0f / B_BAGS) - mean * mean;
        const float sc   = gamma[d] * rsqrtf(var + BN_EPS);
        scale[d] = sc;
        shift[d] = beta[d] - mean * sc;
    }
}

// ---------------------------------------------------------------------------
// 4) Normalize + ReLU + dot with W2 + b2 -> logits. One block per row.
// ---------------------------------------------------------------------------
__global__ __launch_bounds__(256) void bow_head_kernel(
    const float* __restrict__ h, const float* __restrict__ scale,
    const float* __restrict__ shift, const float* __restrict__ W2,
    const float* __restrict__ b2, float* __restrict__ logits)
{
    __shared__ float red[256];
    const int i = blockIdx.x, tid = threadIdx.x;
    const float* hr = h + (size_t)i * D_DIM;
    float p = 0.f;
    for (int d = tid; d < D_DIM; d += 256) {
        const float v = fmaxf(hr[d] * scale[d] + shift[d], 0.0f);
        p += v * W2[d];
    }
    red[tid] = p; __syncthreads();
    for (int off = 128; off; off >>= 1) {
        if (tid < off) red[tid] += red[tid + off];
        __syncthreads();
    }
    if (tid == 0) logits[i] = red[0] + b2[0];
}

// ---------------------------------------------------------------------------
// 5) BCEWithLogits (mean), numerically stable. Single block.
// ---------------------------------------------------------------------------
__global__ __launch_bounds__(256) void bow_loss_kernel(
    const float* __restrict__ logits, const float* __restrict__ t,
    float* __restrict__ loss_out)
{
    __shared__ float red[256];
    const int tid = threadIdx.x;
    float s = 0.f;
    for (int i = tid; i < B_BAGS; i += 256) {
        const float l = logits[i];
        s += fmaxf(l, 0.0f) - l * t[i] + log1pf(expf(-fabsf(l)));
    }
    red[tid] = s; __syncthreads();
    for (int off = 128; off; off >>= 1) {
        if (tid < off) red[tid] += red[tid + off];
        __syncthreads();
    }
    if (tid == 0) loss_out[0] = red[0] * (1.0f / B_BAGS);
}

// ---------------------------------------------------------------------------
extern "C" void kernel_launch(void* const* d_in, const int* in_sizes, int n_in,
                              void* d_out, int out_size, void* d_ws, size_t ws_size,
                              hipStream_t stream) {
    const int*   token_ids = (const int*)  d_in[0];
    const int*   seg       = (const int*)  d_in[1];
    const float* t         = (const float*)d_in[2];
    const float* emb       = (const float*)d_in[3];
    const float* W1        = (const float*)d_in[4];
    const float* b1        = (const float*)d_in[5];
    const float* gamma     = (const float*)d_in[6];
    const float* beta      = (const float*)d_in[7];
    const float* W2        = (const float*)d_in[8];
    const float* b2        = (const float*)d_in[9];
    float*       out       = (float*)d_out;          // [0]=loss, [1..B]=logits
    const int    T         = in_sizes[0];

    float* ws    = (float*)d_ws;
    float* bow   = ws;                                 // B*D
    float* h     = bow + (size_t)B_BAGS * D_DIM;       // B*D
    float* scale = h   + (size_t)B_BAGS * D_DIM;       // D
    float* shift = scale + D_DIM;                      // D

    bow_embed_bag_kernel<<<B_BAGS, 128, 0, stream>>>(token_ids, seg, emb, bow, T);
    bow_gemm_wmma_kernel<<<256, 256, 0, stream>>>(bow, W1, b1, h);
    bow_bn_stats_kernel<<<D_DIM, 256, 0, stream>>>(h, gamma, beta, scale, shift);
    bow_head_kernel<<<B_BAGS, 256, 0, stream>>>(h, scale, shift, W2, b2, out + 1);
    bow_loss_kernel<<<1, 256, 0, stream>>>(out + 1, t, out);
}